// GATConv_dgNN_7370163880453
// MI455X (gfx1250) — compile-verified
//
#include <hip/hip_runtime.h>
#include <math.h>

typedef __attribute__((ext_vector_type(2))) float v2f;
typedef __attribute__((ext_vector_type(8))) float v8f;

#define N_NODES 100000
#define IN_F    64
#define CDIM    256     // OUT*H
#define HEADS   4
#define SLOPE   0.2f

// ---------------------------------------------------------------------------
// K1: Z[N,256] = feat[N,64] @ W^T[64,256] + bias   via V_WMMA_F32_16X16X4_F32
// One wave computes one 16x16 output tile, chaining 16 K-steps of 4.
// Block = 256 threads = 8 waves -> 16 rows x 128 cols per block.
// ---------------------------------------------------------------------------
__global__ __launch_bounds__(256) void gemm_wmma_kernel(
    const float* __restrict__ feat, const float* __restrict__ W,
    const float* __restrict__ bias, float* __restrict__ Z)
{
    const int wave = threadIdx.x >> 5;
    const int lane = threadIdx.x & 31;
    const int half = lane >> 4;          // 0: K 0,1 ; 1: K 2,3
    const int l16  = lane & 15;
    const int rowBase = blockIdx.x * 16;
    const int colBase = blockIdx.y * 128 + wave * 16;

    // A fragment rows: feat[rowBase+l16][*]; B fragment: W[colBase+l16][*]
    const float* arow = feat + (size_t)(rowBase + l16) * IN_F + 2 * half;
    const float* brow = W    + (size_t)(colBase + l16) * IN_F + 2 * half;

    v8f acc = {};
    #pragma unroll
    for (int k0 = 0; k0 < IN_F; k0 += 4) {
        v2f a = *(const v2f*)(arow + k0);   // 8B aligned -> global_load_b64
        v2f b = *(const v2f*)(brow + k0);
        acc = __builtin_amdgcn_wmma_f32_16x16x4_f32(
                  false, a, false, b, (short)0, acc, false, false);
    }
    const float bv = bias[colBase + l16];
    #pragma unroll
    for (int r = 0; r < 8; r++) {
        // C/D layout: VGPR r -> M = r + 8*half, N = l16
        Z[(size_t)(rowBase + r + 8 * half) * CDIM + colBase + l16] = acc[r] + bv;
    }
}

// ---------------------------------------------------------------------------
// K2: per-node attention logits  e_l[i,h] = sum_o Z[i,o*4+h]*a_l[o*4+h]
// One thread per (node, head).
// ---------------------------------------------------------------------------
__global__ __launch_bounds__(256) void logits_kernel(
    const float* __restrict__ Z, const float* __restrict__ a_l,
    const float* __restrict__ a_r, float* __restrict__ el,
    float* __restrict__ er)
{
    const int t = blockIdx.x * blockDim.x + threadIdx.x;
    if (t >= N_NODES * HEADS) return;
    const int i = t >> 2, h = t & 3;
    const float* z = Z + (size_t)i * CDIM + h;
    float sl = 0.f, sr = 0.f;
    #pragma unroll 8
    for (int o = 0; o < 64; o++) {
        const float zv = z[o * 4];
        sl += zv * a_l[o * 4 + h];
        sr += zv * a_r[o * 4 + h];
    }
    el[t] = sl;
    er[t] = sr;
}

// ---------------------------------------------------------------------------
// CSR build: zero counts -> histogram -> exclusive scan -> scatter col ids
// ---------------------------------------------------------------------------
__global__ void zero_kernel(int* __restrict__ p, int n)
{
    const int i = blockIdx.x * blockDim.x + threadIdx.x;
    if (i < n) p[i] = 0;
}

__global__ void hist_kernel(const int* __restrict__ row, int* __restrict__ cnt, int E)
{
    const int e = blockIdx.x * blockDim.x + threadIdx.x;
    if (e < E) atomicAdd(&cnt[row[e]], 1);
}

// Single-block chunked exclusive scan over N counts; writes off[0..N] and fill.
__global__ __launch_bounds__(1024) void scan_kernel(
    const int* __restrict__ cnt, int* __restrict__ off,
    int* __restrict__ fill, int n)
{
    __shared__ int sm[1024];
    __shared__ int carry_s;
    if (threadIdx.x == 0) carry_s = 0;
    __syncthreads();
    for (int base = 0; base < n; base += 1024) {
        const int i = base + (int)threadIdx.x;
        const int v = (i < n) ? cnt[i] : 0;
        sm[threadIdx.x] = v;
        __syncthreads();
        // Hillis-Steele inclusive scan
        for (int d = 1; d < 1024; d <<= 1) {
            const int t = (threadIdx.x >= (unsigned)d) ? sm[threadIdx.x - d] : 0;
            __syncthreads();
            sm[threadIdx.x] += t;
            __syncthreads();
        }
        const int incl  = sm[threadIdx.x];
        const int carry = carry_s;
        if (i < n) {
            const int ex = carry + incl - v;
            off[i]  = ex;
            fill[i] = ex;
        }
        __syncthreads();
        if (threadIdx.x == 1023) carry_s = carry + incl;
        __syncthreads();
    }
    if (threadIdx.x == 0) off[n] = carry_s;
}

__global__ void scatter_kernel(const int* __restrict__ row, const int* __restrict__ col,
                               int* __restrict__ fill, int* __restrict__ ecol, int E)
{
    const int e = blockIdx.x * blockDim.x + threadIdx.x;
    if (e < E) {
        const int pos = atomicAdd(&fill[row[e]], 1);
        ecol[pos] = col[e];   // store source node directly (kills one indirection)
    }
}

// ---------------------------------------------------------------------------
// K3: per-destination softmax + aggregation. One wave32 per node.
// max & sum via __shfl_xor butterflies; output row (256 f32) accumulated in
// registers (8/lane), stored with two B128 stores. No output atomics.
// ---------------------------------------------------------------------------
__global__ __launch_bounds__(256) void aggregate_kernel(
    const float* __restrict__ Z, const float* __restrict__ el,
    const float* __restrict__ er, const int* __restrict__ off,
    const int* __restrict__ ecol, float* __restrict__ out)
{
    const int node = blockIdx.x * 8 + (int)(threadIdx.x >> 5);
    const int lane = threadIdx.x & 31;
    if (node >= N_NODES) return;

    const int beg = off[node];
    const int deg = off[node + 1] - beg;

    float acc[8] = {0.f, 0.f, 0.f, 0.f, 0.f, 0.f, 0.f, 0.f};

    if (deg > 0) {
        const float e0 = el[node * 4 + 0], e1 = el[node * 4 + 1];
        const float e2 = el[node * 4 + 2], e3 = el[node * 4 + 3];

        // pass 1: per-head max over incoming edges
        const float NI = -__builtin_inff();
        float m0 = NI, m1 = NI, m2 = NI, m3 = NI;
        for (int j = lane; j < deg; j += 32) {
            const int c = ecol[beg + j];
            float s0 = e0 + er[c * 4 + 0]; s0 = s0 > 0.f ? s0 : SLOPE * s0;
            float s1 = e1 + er[c * 4 + 1]; s1 = s1 > 0.f ? s1 : SLOPE * s1;
            float s2 = e2 + er[c * 4 + 2]; s2 = s2 > 0.f ? s2 : SLOPE * s2;
            float s3 = e3 + er[c * 4 + 3]; s3 = s3 > 0.f ? s3 : SLOPE * s3;
            m0 = fmaxf(m0, s0); m1 = fmaxf(m1, s1);
            m2 = fmaxf(m2, s2); m3 = fmaxf(m3, s3);
        }
        #pragma unroll
        for (int d = 16; d >= 1; d >>= 1) {
            m0 = fmaxf(m0, __shfl_xor(m0, d, 32));
            m1 = fmaxf(m1, __shfl_xor(m1, d, 32));
            m2 = fmaxf(m2, __shfl_xor(m2, d, 32));
            m3 = fmaxf(m3, __shfl_xor(m3, d, 32));
        }

        // pass 2: per-head sum of exp(a - m)
        float q0 = 0.f, q1 = 0.f, q2 = 0.f, q3 = 0.f;
        for (int j = lane; j < deg; j += 32) {
            const int c = ecol[beg + j];
            float s0 = e0 + er[c * 4 + 0]; s0 = s0 > 0.f ? s0 : SLOPE * s0;
            float s1 = e1 + er[c * 4 + 1]; s1 = s1 > 0.f ? s1 : SLOPE * s1;
            float s2 = e2 + er[c * 4 + 2]; s2 = s2 > 0.f ? s2 : SLOPE * s2;
            float s3 = e3 + er[c * 4 + 3]; s3 = s3 > 0.f ? s3 : SLOPE * s3;
            q0 += expf(s0 - m0); q1 += expf(s1 - m1);
            q2 += expf(s2 - m2); q3 += expf(s3 - m3);
        }
        #pragma unroll
        for (int d = 16; d >= 1; d >>= 1) {
            q0 += __shfl_xor(q0, d, 32);
            q1 += __shfl_xor(q1, d, 32);
            q2 += __shfl_xor(q2, d, 32);
            q3 += __shfl_xor(q3, d, 32);
        }
        const float r0 = 1.f / q0, r1 = 1.f / q1, r2 = 1.f / q2, r3 = 1.f / q3;

        // pass 3: aggregate. All lanes walk edges together; lane covers
        // elements [lane*8, lane*8+8) of the 256-wide row (h = k & 3).
        for (int j = 0; j < deg; j++) {
            const int c = ecol[beg + j];
            float s0 = e0 + er[c * 4 + 0]; s0 = s0 > 0.f ? s0 : SLOPE * s0;
            float s1 = e1 + er[c * 4 + 1]; s1 = s1 > 0.f ? s1 : SLOPE * s1;
            float s2 = e2 + er[c * 4 + 2]; s2 = s2 > 0.f ? s2 : SLOPE * s2;
            float s3 = e3 + er[c * 4 + 3]; s3 = s3 > 0.f ? s3 : SLOPE * s3;
            const float a0 = expf(s0 - m0) * r0;
            const float a1 = expf(s1 - m1) * r1;
            const float a2 = expf(s2 - m2) * r2;
            const float a3 = expf(s3 - m3) * r3;
            const float4* zp = (const float4*)(Z + (size_t)c * CDIM + lane * 8);
            const float4 z0 = zp[0], z1 = zp[1];   // two B128 coalesced gathers
            acc[0] += a0 * z0.x; acc[1] += a1 * z0.y;
            acc[2] += a2 * z0.z; acc[3] += a3 * z0.w;
            acc[4] += a0 * z1.x; acc[5] += a1 * z1.y;
            acc[6] += a2 * z1.z; acc[7] += a3 * z1.w;
        }
    }

    float4* op = (float4*)(out + (size_t)node * CDIM + lane * 8);
    op[0] = make_float4(acc[0], acc[1], acc[2], acc[3]);
    op[1] = make_float4(acc[4], acc[5], acc[6], acc[7]);
}

// ---------------------------------------------------------------------------
extern "C" void kernel_launch(void* const* d_in, const int* in_sizes, int n_in,
                              void* d_out, int out_size, void* d_ws, size_t ws_size,
                              hipStream_t stream)
{
    const float* feat = (const float*)d_in[0];
    const float* Ww   = (const float*)d_in[1];
    const float* Wb   = (const float*)d_in[2];
    const float* a_l  = (const float*)d_in[3];
    const float* a_r  = (const float*)d_in[4];
    const int*   row  = (const int*)d_in[5];
    const int*   col  = (const int*)d_in[6];
    const int    E    = in_sizes[5];

    // workspace layout (~113 MB)
    float* Z    = (float*)d_ws;                         // N*256
    float* el   = Z  + (size_t)N_NODES * CDIM;          // N*4
    float* er   = el + (size_t)N_NODES * HEADS;         // N*4
    int*   cnt  = (int*)(er + (size_t)N_NODES * HEADS); // N
    int*   off  = cnt + N_NODES;                        // N+1
    int*   fill = off + N_NODES + 1;                    // N
    int*   ecol = fill + N_NODES;                       // E

    zero_kernel<<<(N_NODES + 255) / 256, 256, 0, stream>>>(cnt, N_NODES);

    gemm_wmma_kernel<<<dim3(N_NODES / 16, 2), 256, 0, stream>>>(feat, Ww, Wb, Z);

    logits_kernel<<<(N_NODES * HEADS + 255) / 256, 256, 0, stream>>>(Z, a_l, a_r, el, er);

    hist_kernel<<<(E + 255) / 256, 256, 0, stream>>>(row, cnt, E);

    scan_kernel<<<1, 1024, 0, stream>>>(cnt, off, fill, N_NODES);

    scatter_kernel<<<(E + 255) / 256, 256, 0, stream>>>(row, col, fill, ecol, E);

    aggregate_kernel<<<(N_NODES + 7) / 8, 256, 0, stream>>>(Z, el, er, off, ecol,
                                                            (float*)d_out);
}